// OutConv_lstm_5652176962290
// MI455X (gfx1250) — compile-verified
//
#include <hip/hip_runtime.h>
#include <hip/hip_bf16.h>

#define LSEQ 60000
#define NB 8
#define INCH 64
#define HID 5

typedef __attribute__((ext_vector_type(2))) float v2f;
typedef __attribute__((ext_vector_type(8))) float v8f;

// ---------------------------------------------------------------------------
// Kernel 1: 1x1 conv over 64 channels via V_WMMA_F32_16X16X4_F32.
// One wave handles a tile of 16 consecutive l-positions for one batch b.
// A (16x4 f32): lane m (0..15) holds {x[ch k0], x[ch k0+1]} at l=lbase+m,
//               lane m+16 holds {x[ch k0+2], x[ch k0+3]}  (ISA A-layout).
// B (4x16 f32): conv weight per K, replicated across all N columns, so the
//               exact N-lane striping is irrelevant; every column of D equals y.
// D column 0: m=0..7 -> VGPR m lane 0 ; m=8..15 -> VGPR m-8 lane 16.
// ---------------------------------------------------------------------------
__global__ void conv1x1_wmma(const float* __restrict__ x,
                             const float* __restrict__ w,
                             const float* __restrict__ bias,
                             float* __restrict__ y) {
    const int lane = threadIdx.x & 31;
    const int wave = threadIdx.x >> 5;
    const int tile = blockIdx.x * 8 + wave;          // 30000 tiles total
    const int b = tile / (LSEQ / 16);
    const int lbase = (tile % (LSEQ / 16)) * 16;
    const int m = lane & 15;
    const int half = lane >> 4;

    const float* xb = x + (size_t)b * INCH * LSEQ + lbase + m;

    v8f acc = {};
#pragma unroll
    for (int chunk = 0; chunk < 16; ++chunk) {
        const int k0 = chunk * 4 + 2 * half;
        v2f a, bb;
        a.x = xb[(size_t)k0 * LSEQ];
        a.y = xb[(size_t)(k0 + 1) * LSEQ];
        bb.x = w[k0];
        bb.y = w[k0 + 1];
        acc = __builtin_amdgcn_wmma_f32_16x16x4_f32(
            /*neg_a=*/false, a, /*neg_b=*/false, bb,
            /*c_mod=*/(short)0, acc, /*reuse_a=*/false, /*reuse_b=*/false);
    }

    const float bv = bias[0];
    float* yo = y + (size_t)b * LSEQ + lbase;
    if (lane == 0) {
#pragma unroll
        for (int r = 0; r < 8; ++r) yo[r] = acc[r] + bv;
    } else if (lane == 16) {
#pragma unroll
        for (int r = 0; r < 8; ++r) yo[8 + r] = acc[r] + bv;
    }
}

// ---------------------------------------------------------------------------
// Kernel 2: 16 independent projected-LSTM recurrences (2 dirs x 8 batch),
// one thread each; weights live in VGPRs, activations on the trans pipe.
// ---------------------------------------------------------------------------
__device__ __forceinline__ float sigm(float v) {
    return __builtin_amdgcn_rcpf(1.0f + __expf(-v));
}
__device__ __forceinline__ float tanh_fast(float v) {
    // stable for large |v|: exp->inf gives 1, exp->0 gives -1
    return 1.0f - 2.0f * __builtin_amdgcn_rcpf(__expf(2.0f * v) + 1.0f);
}

__global__ void lstm_scan(const float* __restrict__ y,
                          const float* __restrict__ inp,
                          const float* __restrict__ w_ih_f, const float* __restrict__ w_hh_f,
                          const float* __restrict__ b_ih_f, const float* __restrict__ b_hh_f,
                          const float* __restrict__ w_hr_f,
                          const float* __restrict__ w_ih_b, const float* __restrict__ w_hh_b,
                          const float* __restrict__ b_ih_b, const float* __restrict__ b_hh_b,
                          const float* __restrict__ w_hr_b,
                          const float* __restrict__ h0, const float* __restrict__ c0,
                          float* __restrict__ hout) {
    const int t = threadIdx.x;
    if (t >= 16) return;
    const int dir = t >> 3;
    const int b = t & 7;

    const float* w_ih = dir ? w_ih_b : w_ih_f;
    const float* w_hh = dir ? w_hh_b : w_hh_f;
    const float* b_ih = dir ? b_ih_b : b_ih_f;
    const float* b_hh = dir ? b_hh_b : b_hh_f;
    const float* w_hr = dir ? w_hr_b : w_hr_f;

    float wi0[20], wi1[20], whh[20], bsum[20], whr[HID], c[HID];
#pragma unroll
    for (int j = 0; j < 20; ++j) {
        wi0[j] = w_ih[2 * j];
        wi1[j] = w_ih[2 * j + 1];
        whh[j] = w_hh[j];
        bsum[j] = b_ih[j] + b_hh[j];
    }
#pragma unroll
    for (int k = 0; k < HID; ++k) {
        whr[k] = w_hr[k];
        c[k] = c0[(dir * NB + b) * HID + k];
    }
    float h = h0[dir * NB + b];

    const float* yb = y + (size_t)b * LSEQ;
    const float* ib = inp + (size_t)b * LSEQ;
    float* ho = hout + (size_t)(dir * NB + b) * LSEQ;

    for (int s = 0; s < LSEQ; ++s) {
        const int l = dir ? (LSEQ - 1 - s) : s;
        if ((s & 31) == 0) {
            const int pl = dir ? (l - 96) : (l + 96);
            if ((unsigned)pl < (unsigned)LSEQ) {
                __builtin_prefetch(yb + pl, 0, 0);
                __builtin_prefetch(ib + pl, 0, 0);
            }
        }
        const float x0 = yb[l];
        const float x1 = ib[l];
        float hn = 0.0f;
#pragma unroll
        for (int k = 0; k < HID; ++k) {
            const float gi = fmaf(x0, wi0[k],      fmaf(x1, wi1[k],      fmaf(h, whh[k],      bsum[k])));
            const float gf = fmaf(x0, wi0[k + 5],  fmaf(x1, wi1[k + 5],  fmaf(h, whh[k + 5],  bsum[k + 5])));
            const float gc = fmaf(x0, wi0[k + 10], fmaf(x1, wi1[k + 10], fmaf(h, whh[k + 10], bsum[k + 10])));
            const float go = fmaf(x0, wi0[k + 15], fmaf(x1, wi1[k + 15], fmaf(h, whh[k + 15], bsum[k + 15])));
            const float iv = sigm(gi);
            const float fv = sigm(gf);
            const float gv = tanh_fast(gc);
            const float ov = sigm(go);
            c[k] = fmaf(fv, c[k], iv * gv);
            hn = fmaf(ov * tanh_fast(c[k]), whr[k], hn);
        }
        h = hn;
        ho[l] = h;
    }
}

// ---------------------------------------------------------------------------
// Kernel 3: softmax over the sequence axis. softmax(ss) == softmax(ss-1)
// (shift invariance), so both output channels get the same value.
// One block per batch row.
// ---------------------------------------------------------------------------
__global__ void seq_softmax(const float* __restrict__ hbuf, float* __restrict__ out) {
    const int b = blockIdx.x;
    const int tid = threadIdx.x;
    const int T = 256;
    __shared__ float red[256];

    const float* hf = hbuf + (size_t)b * LSEQ;
    const float* hb = hbuf + (size_t)(NB + b) * LSEQ;

    float m = -3.0e38f;
    for (int l = tid; l < LSEQ; l += T) m = fmaxf(m, hf[l] + hb[l]);
    red[tid] = m;
    __syncthreads();
    for (int s = T / 2; s > 0; s >>= 1) {
        if (tid < s) red[tid] = fmaxf(red[tid], red[tid + s]);
        __syncthreads();
    }
    m = red[0];
    __syncthreads();

    float sum = 0.0f;
    for (int l = tid; l < LSEQ; l += T) sum += __expf(hf[l] + hb[l] - m);
    red[tid] = sum;
    __syncthreads();
    for (int s = T / 2; s > 0; s >>= 1) {
        if (tid < s) red[tid] += red[tid + s];
        __syncthreads();
    }
    sum = red[0];
    __syncthreads();

    const float inv = __builtin_amdgcn_rcpf(sum);
    for (int l = tid; l < LSEQ; l += T) {
        const float e = __expf(hf[l] + hb[l] - m) * inv;
        float* o = out + ((size_t)b * LSEQ + l) * 2;
        o[0] = e;
        o[1] = e;
    }
}

extern "C" void kernel_launch(void* const* d_in, const int* in_sizes, int n_in,
                              void* d_out, int out_size, void* d_ws, size_t ws_size,
                              hipStream_t stream) {
    const float* x      = (const float*)d_in[0];
    const float* inp    = (const float*)d_in[1];
    const float* conv_w = (const float*)d_in[2];
    const float* conv_b = (const float*)d_in[3];
    const float* w_ih_f = (const float*)d_in[4];
    const float* w_hh_f = (const float*)d_in[5];
    const float* b_ih_f = (const float*)d_in[6];
    const float* b_hh_f = (const float*)d_in[7];
    const float* w_hr_f = (const float*)d_in[8];
    const float* w_ih_b = (const float*)d_in[9];
    const float* w_hh_b = (const float*)d_in[10];
    const float* b_ih_b = (const float*)d_in[11];
    const float* b_hh_b = (const float*)d_in[12];
    const float* w_hr_b = (const float*)d_in[13];
    const float* h0     = (const float*)d_in[14];
    const float* c0     = (const float*)d_in[15];
    float* out = (float*)d_out;

    float* y    = (float*)d_ws;                 // [B, L]
    float* hbuf = y + (size_t)NB * LSEQ;        // [2, B, L]

    // 30000 wave-tiles, 8 waves per 256-thread block -> 3750 blocks
    conv1x1_wmma<<<(NB * LSEQ / 16) / 8, 256, 0, stream>>>(x, conv_w, conv_b, y);

    lstm_scan<<<1, 32, 0, stream>>>(y, inp,
                                    w_ih_f, w_hh_f, b_ih_f, b_hh_f, w_hr_f,
                                    w_ih_b, w_hh_b, b_ih_b, b_hh_b, w_hr_b,
                                    h0, c0, hbuf);

    seq_softmax<<<NB, 256, 0, stream>>>(hbuf, out);
}